// TopKMIL_85306640433894
// MI455X (gfx1250) — compile-verified
//
#include <hip/hip_runtime.h>

// Problem constants (match reference)
#define B_ 32
#define N_ 4096
#define D_ 768
#define C_ 10
#define K_ 16

typedef __attribute__((ext_vector_type(2))) float v2f;
typedef __attribute__((ext_vector_type(4))) float v4f;
typedef __attribute__((ext_vector_type(8))) float v8f;

// ---------------------------------------------------------------------------
// Kernel 1: scores s[b,n] = dot(H[b,n,:], w) + b_score
// Bandwidth-bound (403 MB of H @ 23.3 TB/s ~= 17 us). One row per wave32,
// 16-byte (global_load_b128) loads per lane, cross-lane shuffle reduction.
// H loads are non-temporal: 403 MB streamed once vs 192 MB L2 — keep it from
// evicting the small reused working set (w_score, scores, W_cls).
// ---------------------------------------------------------------------------
__global__ __launch_bounds__(256) void score_kernel(
    const float* __restrict__ H, const float* __restrict__ w,
    const float* __restrict__ b_score, float* __restrict__ s) {
  const int wave = threadIdx.x >> 5;
  const int lane = threadIdx.x & 31;
  const long row = (long)blockIdx.x * 8 + wave;  // < B*N (exact multiple)
  const v4f* h4 = (const v4f*)(H + row * D_);    // 768 floats, 16B aligned
  const v4f* w4 = (const v4f*)w;
  float acc = 0.f;
#pragma unroll
  for (int j = 0; j < (D_ / 4) / 32; ++j) {  // 6 x 16B per lane
    const int i = j * 32 + lane;
    const v4f hv = __builtin_nontemporal_load(&h4[i]);
    const v4f wv = w4[i];
    acc += hv.x * wv.x + hv.y * wv.y + hv.z * wv.z + hv.w * wv.w;
  }
#pragma unroll
  for (int off = 16; off; off >>= 1) acc += __shfl_xor(acc, off, 32);
  if (lane == 0) s[row] = acc + b_score[0];
}

// ---------------------------------------------------------------------------
// Kernel 2: zero the attention-mask region of d_out (harness poisons it).
// ---------------------------------------------------------------------------
__global__ void zeroA_kernel(float* __restrict__ A) {
  A[(size_t)blockIdx.x * 256 + threadIdx.x] = 0.f;
}

// ---------------------------------------------------------------------------
// Kernel 3: per-batch top-16 by sequential max extraction over LDS keys.
// Key = (orderedFloatBits << 32) | (0xFFFFFFFF - n): max() picks the largest
// score; on ties the LOWER index wins (matches jax.lax.top_k tie-breaking).
// Also scatters A[b,n] = 1/16 at selected indices (16 distinct indices, so the
// reference's normalization is exactly 1/16 per selected slot).
// ---------------------------------------------------------------------------
__global__ __launch_bounds__(256) void topk_kernel(
    const float* __restrict__ s, int* __restrict__ topk, float* __restrict__ A) {
  __shared__ unsigned long long keys[N_];   // 32 KB of the 320 KB LDS
  __shared__ unsigned long long red[256];
  const int b = blockIdx.x;
  const int tid = threadIdx.x;
  for (int n = tid; n < N_; n += 256) {
    unsigned u = __float_as_uint(s[b * N_ + n]);
    u = (u & 0x80000000u) ? ~u : (u | 0x80000000u);  // total order on floats
    keys[n] = ((unsigned long long)u << 32) |
              (unsigned long long)(0xFFFFFFFFu - (unsigned)n);
  }
  __syncthreads();
  for (int it = 0; it < K_; ++it) {
    unsigned long long m = 0ull;
    for (int n = tid; n < N_; n += 256) {
      const unsigned long long k = keys[n];
      m = (k > m) ? k : m;
    }
    red[tid] = m;
    __syncthreads();
    for (int off = 128; off; off >>= 1) {
      if (tid < off) {
        const unsigned long long o = red[tid + off];
        if (o > red[tid]) red[tid] = o;
      }
      __syncthreads();
    }
    if (tid == 0) {
      const unsigned long long best = red[0];
      const int n = (int)(0xFFFFFFFFu - (unsigned)(best & 0xFFFFFFFFull));
      topk[b * K_ + it] = n;
      A[b * N_ + n] = 1.0f / (float)K_;
      keys[n] = 0ull;  // remove from further rounds
    }
    __syncthreads();
  }
}

// ---------------------------------------------------------------------------
// Kernel 4: M[b,:] = mean of the 16 gathered rows of H (order-invariant).
// ---------------------------------------------------------------------------
__global__ __launch_bounds__(256) void gather_mean_kernel(
    const float* __restrict__ H, const int* __restrict__ topk,
    float* __restrict__ M) {
  __shared__ int idx[K_];
  const int b = blockIdx.x;
  if (threadIdx.x < K_) idx[threadIdx.x] = topk[b * K_ + threadIdx.x];
  __syncthreads();
  for (int d = threadIdx.x; d < D_; d += 256) {
    float acc = 0.f;
#pragma unroll
    for (int j = 0; j < K_; ++j)
      acc += H[((long)b * N_ + idx[j]) * D_ + d];
    M[b * D_ + d] = acc * (1.0f / (float)K_);
  }
}

// ---------------------------------------------------------------------------
// Kernel 5: logits = M[32,768] @ W_cls[768,10] + b_cls via V_WMMA_F32_16X16X4_F32.
// Two waves, each owns a 16x16 output tile (rows 0-15 / 16-31), K stepped by 4.
// W_cls is zero-padded to 16 columns and staged in LDS ONCE, so the WMMA inner
// loop has no divergent predicated loads (EXEC stays all-ones as WMMA needs;
// no per-iteration saveexec / restore around the loads).
// f32 A-fragment layout (ISA 7.12.2): 2 VGPRs; VGPR0 = K {0|2} for lane halves,
// VGPR1 = K {1|3}. C/D layout: VGPR r -> row r (lanes 0-15) / row r+8 (16-31).
// ---------------------------------------------------------------------------
__global__ __launch_bounds__(64) void logits_wmma_kernel(
    const float* __restrict__ M, const float* __restrict__ Wc,
    const float* __restrict__ b_cls, float* __restrict__ logits) {
  __shared__ float Wpad[D_ * 16];  // 48 KB: W_cls padded [768][16]
  const int tid = threadIdx.x;
  for (int i = tid; i < D_ * 16; i += 64) {
    const int krow = i >> 4;
    const int col = i & 15;
    Wpad[i] = (col < C_) ? Wc[krow * C_ + col] : 0.f;
  }
  __syncthreads();

  const int wave = tid >> 5;
  const int lane = tid & 31;
  const int half = lane >> 4;   // which 16-lane half
  const int l16 = lane & 15;
  const int rowbase = wave * 16;

  v8f c = {};
  for (int k = 0; k < D_; k += 4) {
    // A fragment: row = rowbase + l16; VGPR0 -> K = k + 2*half, VGPR1 -> +1
    const float* mrow = M + (size_t)(rowbase + l16) * D_ + k + half * 2;
    v2f a;
    a.x = mrow[0];
    a.y = mrow[1];
    // B fragment from LDS (unconditional): col = l16;
    // VGPR0 -> K = k + half, VGPR1 -> K = k + 2 + half
    v2f bf;
    bf.x = Wpad[(k + half) * 16 + l16];
    bf.y = Wpad[(k + 2 + half) * 16 + l16];
    // 8 args: (neg_a, A, neg_b, B, c_mod, C, reuse_a, reuse_b)
    c = __builtin_amdgcn_wmma_f32_16x16x4_f32(false, a, false, bf,
                                              (short)0, c, false, false);
  }
  const float bias = (l16 < C_) ? b_cls[l16] : 0.f;
#pragma unroll
  for (int r = 0; r < 8; ++r) {
    const int row = rowbase + r + half * 8;
    if (l16 < C_) logits[row * C_ + l16] = c[r] + bias;
  }
}

// ---------------------------------------------------------------------------
// Host-side launcher
// d_in order: H [B*N*D], w_score [D], b_score [1], W_cls [D*C], b_cls [C]
// d_out: logits [B*C] then A [B*N] (flattened tuple, f32)
// d_ws:  s [B*N] f32 | topk [B*K] i32 | M [B*D] f32   (~610 KB)
// ---------------------------------------------------------------------------
extern "C" void kernel_launch(void* const* d_in, const int* in_sizes, int n_in,
                              void* d_out, int out_size, void* d_ws,
                              size_t ws_size, hipStream_t stream) {
  const float* H       = (const float*)d_in[0];
  const float* w_score = (const float*)d_in[1];
  const float* b_score = (const float*)d_in[2];
  const float* W_cls   = (const float*)d_in[3];
  const float* b_cls   = (const float*)d_in[4];

  float* logits = (float*)d_out;           // B*C = 320 floats
  float* A      = logits + B_ * C_;        // B*N = 131072 floats

  float* s    = (float*)d_ws;              // B*N floats
  int*   topk = (int*)(s + B_ * N_);       // B*K ints
  float* M    = (float*)(topk + B_ * K_);  // B*D floats

  // 1) scores: one row per wave32, 8 rows per 256-thread block
  score_kernel<<<(B_ * N_) / 8, 256, 0, stream>>>(H, w_score, b_score, s);
  // 2) zero attention-mask output region
  zeroA_kernel<<<(B_ * N_) / 256, 256, 0, stream>>>(A);
  // 3) per-batch top-16 + scatter 1/16 into A
  topk_kernel<<<B_, 256, 0, stream>>>(s, topk, A);
  // 4) gather + mean
  gather_mean_kernel<<<B_, 256, 0, stream>>>(H, topk, M);
  // 5) classifier via f32 WMMA (LDS-staged zero-padded W)
  logits_wmma_kernel<<<1, 64, 0, stream>>>(M, W_cls, b_cls, logits);
}